// RemapLayer_22376779612836
// MI455X (gfx1250) — compile-verified
//
#include <hip/hip_runtime.h>

typedef float v2f __attribute__((ext_vector_type(2)));
typedef float v4f __attribute__((ext_vector_type(4)));
typedef float v8f __attribute__((ext_vector_type(8)));

#define TPB 256
#define NB_PART 1024
#define NB_MAP 2048

// ---------------------------------------------------------------------------
// Kernel A: streaming partial sum / sum-of-squares, accumulated through
// V_WMMA_F32_16X16X4_F32 (B = ones => D[m][n] += sum_k A[m][k]).
// Sum is permutation invariant, so arbitrary lane->A-slot packing is valid.
// ---------------------------------------------------------------------------
__global__ __launch_bounds__(TPB) void partial_moments_wmma(
    const float* __restrict__ x, long long n, float* __restrict__ part) {
  const int tid    = blockIdx.x * blockDim.x + threadIdx.x;
  const int stride = gridDim.x * blockDim.x;

  const long long chunks = n >> 2;              // float4 chunks
  const long long full   = chunks / stride;     // wave-uniform trip count (EXEC all-1s)

  v8f c_sum = {};
  v8f c_sq  = {};
  const v2f ones = {1.0f, 1.0f};
  const v4f* __restrict__ x4 = (const v4f*)x;

  for (long long j = 0; j < full; ++j) {
    v4f v = x4[j * (long long)stride + tid];
    v2f a0 = {v.x, v.y};
    v2f a1 = {v.z, v.w};
    v2f q0 = {v.x * v.x, v.y * v.y};
    v2f q1 = {v.z * v.z, v.w * v.w};
    c_sum = __builtin_amdgcn_wmma_f32_16x16x4_f32(false, a0, false, ones, (short)0, c_sum, false, false);
    c_sum = __builtin_amdgcn_wmma_f32_16x16x4_f32(false, a1, false, ones, (short)0, c_sum, false, false);
    c_sq  = __builtin_amdgcn_wmma_f32_16x16x4_f32(false, q0, false, ones, (short)0, c_sq,  false, false);
    c_sq  = __builtin_amdgcn_wmma_f32_16x16x4_f32(false, q1, false, ones, (short)0, c_sq,  false, false);
  }

  // Fold C tile: lane holds rows (0-7) or (8-15) replicated across N.
  float ls = 0.0f, lq = 0.0f;
#pragma unroll
  for (int r = 0; r < 8; ++r) { ls += c_sum[r]; lq += c_sq[r]; }
  ls += __shfl_xor(ls, 16, 32);   // add the other half-tile -> full wave total (all lanes)
  lq += __shfl_xor(lq, 16, 32);

  // Scalar tail (no WMMA; divergence allowed here).
  float ts = 0.0f, tq = 0.0f;
  for (long long i = full * (long long)stride * 4 + tid; i < n; i += stride) {
    float v = x[i];
    ts += v;
    tq += v * v;
  }

  // Deterministic block tree-reduction: wave totals (lane 0 only) + tail parts.
  __shared__ float bs[TPB];
  __shared__ float bq[TPB];
  const int lane = threadIdx.x & 31;
  bs[threadIdx.x] = (lane == 0 ? ls : 0.0f) + ts;
  bq[threadIdx.x] = (lane == 0 ? lq : 0.0f) + tq;
  __syncthreads();
  for (int off = TPB / 2; off > 0; off >>= 1) {
    if ((int)threadIdx.x < off) {
      bs[threadIdx.x] += bs[threadIdx.x + off];
      bq[threadIdx.x] += bq[threadIdx.x + off];
    }
    __syncthreads();
  }
  if (threadIdx.x == 0) {
    part[blockIdx.x]           = bs[0];
    part[NB_PART + blockIdx.x] = bq[0];
  }
}

// ---------------------------------------------------------------------------
// Kernel B: reduce NB_PART partials -> mean, 1/std (ddof=1), clamped scale,
// fused constant 127.5/s. Single block, fixed order => deterministic.
// ---------------------------------------------------------------------------
__global__ __launch_bounds__(TPB) void finalize_stats(
    const float* __restrict__ part, const float* __restrict__ scale,
    float* __restrict__ params, long long n) {
  __shared__ float rs[TPB];
  __shared__ float rq[TPB];
  float s = 0.0f, q = 0.0f;
  for (int i = threadIdx.x; i < NB_PART; i += TPB) {
    s += part[i];
    q += part[NB_PART + i];
  }
  rs[threadIdx.x] = s;
  rq[threadIdx.x] = q;
  __syncthreads();
  for (int off = TPB / 2; off > 0; off >>= 1) {
    if ((int)threadIdx.x < off) {
      rs[threadIdx.x] += rs[threadIdx.x + off];
      rq[threadIdx.x] += rq[threadIdx.x + off];
    }
    __syncthreads();
  }
  if (threadIdx.x == 0) {
    float S    = rs[0];
    float Q    = rq[0];
    float mean = S / (float)n;
    float var  = (Q - S * mean) / (float)(n - 1);   // unbiased (ddof=1)
    var        = fmaxf(var, 0.0f);
    float inv_std = 1.0f / sqrtf(var);
    float sc   = fminf(fmaxf(scale[0], 2.5f), 3.5f);
    params[0] = mean;
    params[1] = inv_std;
    params[2] = sc;
    params[3] = 127.5f / sc;
  }
}

// ---------------------------------------------------------------------------
// Kernel C: streaming remap. emb LUT in LDS, b128 loads, NT b128 stores.
// t = clamp((x-mean)*inv_std, -s, s) * (127.5/s) + 127.5   in [0, 255]
// out = frac*emb[floor(t)] + (1-frac)*emb[ceil(t)]   (faithful swapped weights)
// ---------------------------------------------------------------------------
__global__ __launch_bounds__(TPB) void remap_kernel(
    const float* __restrict__ x, const float* __restrict__ emb,
    const float* __restrict__ params, float* __restrict__ out, long long n) {
  __shared__ float lut[256];
  for (int i = threadIdx.x; i < 256; i += TPB) lut[i] = emb[i];
  __syncthreads();

  const float mean    = params[0];
  const float inv_std = params[1];
  const float s       = params[2];
  const float k       = params[3];

  const int tid    = blockIdx.x * blockDim.x + threadIdx.x;
  const int stride = gridDim.x * blockDim.x;
  const long long chunks = n >> 2;

  const v4f* __restrict__ x4 = (const v4f*)x;
  v4f* __restrict__ o4 = (v4f*)out;

  for (long long i = tid; i < chunks; i += stride) {
    v4f v = x4[i];
    v4f r;
#pragma unroll
    for (int c = 0; c < 4; ++c) {
      float z  = (v[c] - mean) * inv_std;
      float cl = fminf(fmaxf(z, -s), s);
      float t  = fmaf(cl, k, 127.5f);
      t        = fminf(fmaxf(t, 0.0f), 255.0f);   // guard ulp spill at the rails
      float fl = floorf(t);
      float cu = ceilf(t);
      float frac = t - fl;
      float lv = lut[(int)fl];
      float uv = lut[(int)cu];
      r[c] = fmaf(frac, lv, (1.0f - frac) * uv);
    }
    __builtin_nontemporal_store(r, o4 + i);       // write-once output: NT b128
  }

  // Scalar tail (n % 4 elements).
  for (long long i = (chunks << 2) + tid; i < n; i += stride) {
    float z  = (x[i] - mean) * inv_std;
    float cl = fminf(fmaxf(z, -s), s);
    float t  = fmaf(cl, k, 127.5f);
    t        = fminf(fmaxf(t, 0.0f), 255.0f);
    float fl = floorf(t);
    float frac = t - fl;
    float lv = lut[(int)fl];
    float uv = lut[(int)ceilf(t)];
    out[i] = fmaf(frac, lv, (1.0f - frac) * uv);
  }
}

extern "C" void kernel_launch(void* const* d_in, const int* in_sizes, int n_in,
                              void* d_out, int out_size, void* d_ws, size_t ws_size,
                              hipStream_t stream) {
  (void)n_in; (void)out_size; (void)ws_size;
  const float* x     = (const float*)d_in[0];
  const float* emb   = (const float*)d_in[1];
  const float* scale = (const float*)d_in[2];
  float* out = (float*)d_out;
  const long long n = (long long)in_sizes[0];

  float* part   = (float*)d_ws;            // NB_PART sums | NB_PART sumsqs
  float* params = part + 2 * NB_PART;      // mean, 1/std, s, 127.5/s

  partial_moments_wmma<<<NB_PART, TPB, 0, stream>>>(x, n, part);
  finalize_stats<<<1, TPB, 0, stream>>>(part, scale, params, n);
  remap_kernel<<<NB_MAP, TPB, 0, stream>>>(x, emb, params, out, n);
}